// CrossAttentionPoseRegression_11742440587710
// MI455X (gfx1250) — compile-verified
//
#include <hip/hip_runtime.h>
#include <math.h>

#define NPTS 524288
#define HFD  33
#define KTOP 128
#define TPB  256
#define NB   (NPTS / TPB)   // 2048 blocks exactly covers N

typedef __attribute__((ext_vector_type(2))) float v2f;
typedef __attribute__((ext_vector_type(8))) float v8f;

// ---- monotone float <-> uint mapping (order-preserving) ----
__device__ __forceinline__ unsigned mapf(float f) {
  unsigned u = __float_as_uint(f);
  return (u & 0x80000000u) ? ~u : (u | 0x80000000u);
}
__device__ __forceinline__ float unmapf(unsigned m) {
  unsigned u = (m & 0x80000000u) ? (m & 0x7FFFFFFFu) : ~m;
  return __uint_as_float(u);
}

// =====================================================================
// init: zero hist + selection state
// =====================================================================
__global__ void capr_init(unsigned* hist, unsigned* sel, unsigned long long* maxkey) {
  int t = threadIdx.x;
  if (t < 256) hist[t] = 0u;
  if (t == 0) {
    sel[0] = 0u;          // radix prefix
    sel[1] = KTOP;        // remaining rank
    sel[2] = 1u;          // topidx fill counter (slot 0 reserved for argmax)
    sel[3] = 0u;          // eq counter
    sel[4] = 0u;          // max mapped (post-scatter)
    sel[5] = 0xFFFFFFFFu; // min mapped (post-scatter)
    *maxkey = 0ull;
  }
}

// =====================================================================
// big coalesced copies: hs, xs, ht, xt -> out (the dominant traffic)
// =====================================================================
__global__ void capr_copy(const float* __restrict__ hs, const float* __restrict__ xs,
                          const float* __restrict__ ht, const float* __restrict__ xt,
                          float* __restrict__ out) {
  const long long NH = (long long)NPTS * HFD;
  const long long N3 = (long long)NPTS * 3;
  long long stride = (long long)gridDim.x * blockDim.x;
  long long g = (long long)blockIdx.x * blockDim.x + threadIdx.x;
  float* o_hs = out + 13;
  float* o_xs = o_hs + NH;
  float* o_ht = o_xs + N3;
  float* o_xt = o_ht + NH;
  for (long long i = g; i < NH; i += stride) { o_hs[i] = hs[i]; o_ht[i] = ht[i]; }
  for (long long i = g; i < N3; i += stride) { o_xs[i] = xs[i]; o_xt[i] = xt[i]; }
}

// =====================================================================
// fused per-point pass: sim, losses (block partials), packed global argmax
// =====================================================================
__global__ __launch_bounds__(TPB) void capr_point(
    const float* __restrict__ hs, const float* __restrict__ ht,
    const float* __restrict__ xs, const float* __restrict__ xt,
    const float* __restrict__ labels, const float* __restrict__ gt,
    float* __restrict__ sim, unsigned long long* maxkey, float* __restrict__ partA) {
  __shared__ float red[TPB];
  __shared__ unsigned long long kred[TPB];
  const int t = threadIdx.x;
  const int i = blockIdx.x * TPB + t;

  const float* a = hs + (long long)i * HFD;
  const float* b = ht + (long long)i * HFD;
  float dot = 0.f, na = 0.f, nb = 0.f;
#pragma unroll
  for (int c = 0; c < HFD; ++c) { float x = a[c], y = b[c]; dot += x * y; na += x * x; nb += y * y; }
  sim[i] = dot;

  // packed (value, ~index) -> argmax with lowest-index tie break
  kred[t] = ((unsigned long long)mapf(dot) << 32) | (unsigned long long)(0xFFFFFFFFu - (unsigned)i);
  __syncthreads();
  for (int o = TPB / 2; o > 0; o >>= 1) {
    if (t < o) kred[t] = kred[t] > kred[t + o] ? kred[t] : kred[t + o];
    __syncthreads();
  }
  if (t == 0) atomicMax(maxkey, kred[0]);

  float lab = labels[i];
  float nrm = fmaxf(sqrtf(na) * sqrtf(nb), 1e-8f);
  float cosv = dot / nrm;
  float feat = (cosv - lab) * (cosv - lab);

  float p0 = xs[(long long)i * 3 + 0], p1 = xs[(long long)i * 3 + 1], p2 = xs[(long long)i * 3 + 2];
  float q0 = xt[(long long)i * 3 + 0], q1 = xt[(long long)i * 3 + 1], q2 = xt[(long long)i * 3 + 2];
  float r0 = gt[0] * p0 + gt[1] * p1 + gt[2] * p2 + gt[3];
  float r1 = gt[4] * p0 + gt[5] * p1 + gt[6] * p2 + gt[7];
  float r2 = gt[8] * p0 + gt[9] * p1 + gt[10] * p2 + gt[11];
  float d0 = r0 - q0, d1 = r1 - q1, d2 = r2 - q2;
  float rot = (d0 * d0 + d1 * d1 + d2 * d2) * lab;

  red[t] = rot; __syncthreads();
  for (int o = TPB / 2; o > 0; o >>= 1) { if (t < o) red[t] += red[t + o]; __syncthreads(); }
  if (t == 0) partA[blockIdx.x * 2 + 0] = red[0];
  __syncthreads();
  red[t] = feat; __syncthreads();
  for (int o = TPB / 2; o > 0; o >>= 1) { if (t < o) red[t] += red[t + o]; __syncthreads(); }
  if (t == 0) partA[blockIdx.x * 2 + 1] = red[0];
}

// =====================================================================
// radix select (MSB-first, 8 bits/pass) for the K-th largest mapped key
// =====================================================================
__global__ void capr_histzero(unsigned* hist) { hist[threadIdx.x] = 0u; }

__global__ __launch_bounds__(TPB) void capr_hist(const float* __restrict__ sim,
                                                 unsigned* hist, const unsigned* sel, int shift) {
  __shared__ unsigned h[256];
  const int t = threadIdx.x;
  h[t] = 0u; __syncthreads();
  const int i = blockIdx.x * TPB + t;
  unsigned key = mapf(sim[i]);
  unsigned maskHigh = (shift >= 24) ? 0u : (0xFFFFFFFFu << (shift + 8));
  if ((key & maskHigh) == sel[0]) atomicAdd(&h[(key >> shift) & 0xFFu], 1u);
  __syncthreads();
  if (h[t]) atomicAdd(&hist[t], h[t]);
}

__global__ void capr_sel(const unsigned* hist, unsigned* sel, unsigned* topidx,
                         const unsigned long long* maxkey, int shift) {
  unsigned rem = sel[1];
  for (int b = 255; b >= 0; --b) {
    unsigned c = hist[b];
    if (rem > c) rem -= c;
    else { sel[0] |= ((unsigned)b) << shift; break; }
  }
  sel[1] = rem;
  if (shift == 0) topidx[0] = 0xFFFFFFFFu - (unsigned)((*maxkey) & 0xFFFFFFFFull);
}

// =====================================================================
// gather strict-greater indices (argmax pinned to slot 0), collect equals
// =====================================================================
__global__ __launch_bounds__(TPB) void capr_gather(const float* __restrict__ sim,
                                                   unsigned* sel, unsigned* topidx, unsigned* eqbuf) {
  const int i = blockIdx.x * TPB + threadIdx.x;
  unsigned key = mapf(sim[i]);
  unsigned thresh = sel[0];
  unsigned amax = topidx[0];
  if ((unsigned)i == amax) return;
  if (key > thresh) {
    unsigned pos = atomicAdd(&sel[2], 1u);
    if (pos < KTOP) topidx[pos] = (unsigned)i;
  } else if (key == thresh) {
    unsigned e = atomicAdd(&sel[3], 1u);
    if (e < 4096u) eqbuf[e] = (unsigned)i;
  }
}

__global__ __launch_bounds__(TPB) void capr_fill(unsigned* sel, unsigned* topidx, unsigned* eqbuf) {
  __shared__ unsigned rv[TPB];
  const int t = threadIdx.x;
  unsigned cnt = sel[2];
  int needed = KTOP - (int)cnt;
  unsigned eqn = sel[3] < 4096u ? sel[3] : 4096u;
  for (int j = 0; j < needed; ++j) {
    unsigned best = 0xFFFFFFFFu;
    for (unsigned i = t; i < eqn; i += TPB) { unsigned v = eqbuf[i]; best = v < best ? v : best; }
    rv[t] = best; __syncthreads();
    for (int o = TPB / 2; o > 0; o >>= 1) { if (t < o) rv[t] = rv[t + o] < rv[t] ? rv[t + o] : rv[t]; __syncthreads(); }
    unsigned m = rv[0];
    if (t == 0 && m != 0xFFFFFFFFu) topidx[cnt + j] = m;
    for (unsigned i = t; i < eqn; i += TPB) if (eqbuf[i] == m) eqbuf[i] = 0xFFFFFFFFu;
    __syncthreads();
  }
}

// =====================================================================
// MLP on the 128 gathered rows with V_WMMA_F32_16X16X4_F32 (f32 in/out).
// ISA layouts (CDNA5 ISA):
//   A 16x4 : lanes 0-15 -> M, lanes 16-31 -> same M with K+2; VGPR0=K, VGPR1=K+1
//   B 4x16 : lanes 0-15 -> N (K rows 0/1 in VGPR0/1), lanes 16-31 -> N (K rows 2/3)
//   C/D    : VGPR g -> row g (+8 for lanes 16-31), col = lane&15
// 8 waves: wave w owns row tile w (m0=16w); layer1 3 N-tiles x 17 K-steps,
// layer2 1 N-tile x 9 K-steps. Shared pool 63.2KB; stage-2 aliases dead X region.
// =====================================================================
__global__ __launch_bounds__(TPB) void capr_mlp(
    const float* __restrict__ hs, const float* __restrict__ ht,
    const float* __restrict__ W1, const float* __restrict__ b1,
    const float* __restrict__ W2, const float* __restrict__ b2,
    const float* __restrict__ W3, const float* __restrict__ b3,
    const unsigned* __restrict__ topidx, const unsigned long long* __restrict__ maxkey,
    float* __restrict__ scal) {
  __shared__ float smem[15800];
#define XS(r, c)  smem[(r) * 68 + (c)]              // 128x68 (K padded to 68)
#define B1S(k, c) smem[8704 + (k) * 34 + (c)]       // 68x34  (W1^T, col 33 zero pad)
#define Y1S(r, c) smem[11016 + (r) * 36 + (c)]      // 128x36 (cols 33..35 zero)
#define B1V(c)    smem[15624 + (c)]                 // 48 bias (zero padded)
#define TISM ((unsigned*)(smem + 15672))            // 128 gathered indices
#define B2S(k, c) smem[(k) * 16 + (c)]              // 36x16  (aliases XS, stage 2)
#define Y2S(r, c) smem[576 + (r) * 16 + (c)]        // 128x16
#define B2V(c)    smem[2624 + (c)]
#define SPRED(r)  smem[2640 + (r)]

  const int tid = threadIdx.x;
  const int lane = tid & 31;
  const int w = tid >> 5;

  if (tid < 128) TISM[tid] = topidx[tid];
  __syncthreads();

  for (int idx = tid; idx < 128 * 66; idx += TPB) {
    int r = idx / 66, c = idx % 66;
    unsigned src = TISM[r];
    XS(r, c) = (c < HFD) ? hs[(size_t)src * HFD + c] : ht[(size_t)src * HFD + (c - HFD)];
  }
  for (int r = tid; r < 128; r += TPB) { XS(r, 66) = 0.f; XS(r, 67) = 0.f; }
  for (int idx = tid; idx < 68 * 34; idx += TPB) {
    int k = idx / 34, c = idx % 34;
    B1S(k, c) = (k < 66 && c < HFD) ? W1[c * 66 + k] : 0.f;
  }
  if (tid < 48) B1V(tid) = (tid < HFD) ? b1[tid] : 0.f;
  __syncthreads();

  const int m0 = w * 16;
  const int mrow = m0 + (lane & 15);
  const int khalf = (lane >> 4) << 1;

  // ---- layer 1: Y1 = relu(X @ W1^T + b1), 128x66 -> 128x33 ----
  for (int nt = 0; nt < 3; ++nt) {
    const int colr = nt * 16 + (lane & 15);
    float bias = B1V(colr);
    v8f c;
#pragma unroll
    for (int g = 0; g < 8; ++g) c[g] = bias;
#pragma unroll
    for (int s = 0; s < 17; ++s) {
      int kk = s * 4 + khalf;
      v2f a, b;
      a.x = XS(mrow, kk); a.y = XS(mrow, kk + 1);
      b.x = (colr < 34) ? B1S(kk, colr) : 0.f;
      b.y = (colr < 34) ? B1S(kk + 1, colr) : 0.f;
      c = __builtin_amdgcn_wmma_f32_16x16x4_f32(false, a, false, b, (short)0, c, false, false);
    }
    const int rbase = m0 + ((lane >> 4) << 3);
    if (colr < 36) {
#pragma unroll
      for (int g = 0; g < 8; ++g) Y1S(rbase + g, colr) = fmaxf(c[g], 0.f);
    }
  }
  __syncthreads();

  for (int idx = tid; idx < 36 * 16; idx += TPB) {
    int k = idx / 16, c = idx % 16;
    B2S(k, c) = (k < HFD) ? W2[c * HFD + k] : 0.f;
  }
  if (tid < 16) B2V(tid) = b2[tid];
  __syncthreads();

  // ---- layer 2: Y2 = relu(Y1 @ W2^T + b2), 128x33 -> 128x16 ----
  {
    const int colr = lane & 15;
    float bias = B2V(colr);
    v8f c;
#pragma unroll
    for (int g = 0; g < 8; ++g) c[g] = bias;
#pragma unroll
    for (int s = 0; s < 9; ++s) {
      int kk = s * 4 + khalf;
      v2f a, b;
      a.x = Y1S(mrow, kk); a.y = Y1S(mrow, kk + 1);
      b.x = B2S(kk, colr); b.y = B2S(kk + 1, colr);
      c = __builtin_amdgcn_wmma_f32_16x16x4_f32(false, a, false, b, (short)0, c, false, false);
    }
    const int rbase = m0 + ((lane >> 4) << 3);
#pragma unroll
    for (int g = 0; g < 8; ++g) Y2S(rbase + g, colr) = fmaxf(c[g], 0.f);
  }
  __syncthreads();

  // ---- layer 3 + scalar replace logic ----
  if (tid < 128) {
    float acc = b3[0];
#pragma unroll
    for (int c = 0; c < 16; ++c) acc += Y2S(tid, c) * W3[c];
    SPRED(tid) = acc * (1.0f / (float)KTOP);
  }
  __syncthreads();
  if (tid == 0) {
    float p0 = SPRED(0);                                   // row 0 == argmax row
    float tkv = unmapf((unsigned)((*maxkey) >> 32));       // top_scores[0]
    bool cond = (p0 > 0.5f) && ((fabsf(p0 - 1.0f) < tkv) || (p0 < tkv));
    scal[0] = cond ? p0 : tkv;                             // replaced
  }
#undef XS
#undef B1S
#undef Y1S
#undef B1V
#undef TISM
#undef B2S
#undef Y2S
#undef B2V
#undef SPRED
}

// =====================================================================
// scatter replaced value into sim at the top-K index set
// =====================================================================
__global__ void capr_scatter(float* sim, const unsigned* topidx, const float* scal) {
  sim[topidx[threadIdx.x]] = scal[0];
}

// =====================================================================
// sum + min/max of modified sim (block partials, integer min/max atomics)
// =====================================================================
__global__ __launch_bounds__(TPB) void capr_sum1(const float* __restrict__ sim,
                                                 float* __restrict__ partS, unsigned* sel) {
  __shared__ float red[TPB];
  __shared__ unsigned ur[TPB];
  const int t = threadIdx.x;
  const int i = blockIdx.x * TPB + t;
  float v = sim[i];
  unsigned m = mapf(v);
  red[t] = v; ur[t] = m; __syncthreads();
  for (int o = TPB / 2; o > 0; o >>= 1) {
    if (t < o) { red[t] += red[t + o]; ur[t] = ur[t + o] > ur[t] ? ur[t + o] : ur[t]; }
    __syncthreads();
  }
  if (t == 0) { partS[blockIdx.x] = red[0]; atomicMax(&sel[4], ur[0]); }
  __syncthreads();
  ur[t] = m; __syncthreads();
  for (int o = TPB / 2; o > 0; o >>= 1) { if (t < o) ur[t] = ur[t + o] < ur[t] ? ur[t + o] : ur[t]; __syncthreads(); }
  if (t == 0) atomicMin(&sel[5], ur[0]);
}

__global__ __launch_bounds__(TPB) void capr_prep(const float* __restrict__ partS,
                                                 const unsigned* sel, float* scal) {
  __shared__ float red[TPB];
  const int t = threadIdx.x;
  float s = 0.f;
  for (int i = t; i < NB; i += TPB) s += partS[i];
  red[t] = s; __syncthreads();
  for (int o = TPB / 2; o > 0; o >>= 1) { if (t < o) red[t] += red[t + o]; __syncthreads(); }
  if (t == 0) {
    float S = red[0] + 1e-6f;                       // fw denominator
    float mx = unmapf(sel[4]), mn = unmapf(sel[5]);
    scal[1] = S;
    scal[2] = (S > 0.f) ? mx / S : mn / S;          // max(fw) for softmax
  }
}

// =====================================================================
// softmax-weighted sums: [sum e, e*ps(3), e*pt(3), e*ps_j*pt_k(9)] partials
// =====================================================================
__global__ __launch_bounds__(TPB) void capr_wsum(const float* __restrict__ sim,
                                                 const float* __restrict__ xs,
                                                 const float* __restrict__ xt,
                                                 const float* __restrict__ scal,
                                                 float* __restrict__ part16) {
  __shared__ float red[TPB];
  const int t = threadIdx.x;
  const int i = blockIdx.x * TPB + t;
  float S = scal[1], fwmax = scal[2];
  float e = expf(sim[i] / S - fwmax);
  float p[3] = { xs[(long long)i * 3 + 0], xs[(long long)i * 3 + 1], xs[(long long)i * 3 + 2] };
  float q[3] = { xt[(long long)i * 3 + 0], xt[(long long)i * 3 + 1], xt[(long long)i * 3 + 2] };
  float vals[16];
  vals[0] = e;
#pragma unroll
  for (int j = 0; j < 3; ++j) { vals[1 + j] = e * p[j]; vals[4 + j] = e * q[j]; }
#pragma unroll
  for (int j = 0; j < 3; ++j)
#pragma unroll
    for (int k = 0; k < 3; ++k) vals[7 + j * 3 + k] = e * p[j] * q[k];
  for (int v = 0; v < 16; ++v) {
    red[t] = vals[v]; __syncthreads();
    for (int o = TPB / 2; o > 0; o >>= 1) { if (t < o) red[t] += red[t + o]; __syncthreads(); }
    if (t == 0) part16[blockIdx.x * 16 + v] = red[0];
    __syncthreads();
  }
}

// =====================================================================
// 3x3 SVD (Jacobi on H^T H) + Kabsch with reference's det sign fix
// =====================================================================
__device__ void capr_kabsch3(const double H[3][3], double R[3][3]) {
  double A[3][3], V[3][3] = { {1, 0, 0}, {0, 1, 0}, {0, 0, 1} };
  for (int i = 0; i < 3; ++i)
    for (int j = 0; j < 3; ++j)
      A[i][j] = H[0][i] * H[0][j] + H[1][i] * H[1][j] + H[2][i] * H[2][j];
  const int P[3] = { 0, 0, 1 }, Q[3] = { 1, 2, 2 };
  for (int sweep = 0; sweep < 40; ++sweep) {
    for (int r = 0; r < 3; ++r) {
      int p = P[r], q = Q[r];
      double apq = A[p][q];
      if (fabs(apq) < 1e-300) continue;
      double tau = (A[q][q] - A[p][p]) / (2.0 * apq);
      double tt = (tau >= 0.0 ? 1.0 : -1.0) / (fabs(tau) + sqrt(1.0 + tau * tau));
      double cth = 1.0 / sqrt(1.0 + tt * tt), sth = tt * cth;
      for (int k = 0; k < 3; ++k) { double a0 = A[k][p], a1 = A[k][q]; A[k][p] = cth * a0 - sth * a1; A[k][q] = sth * a0 + cth * a1; }
      for (int k = 0; k < 3; ++k) { double a0 = A[p][k], a1 = A[q][k]; A[p][k] = cth * a0 - sth * a1; A[q][k] = sth * a0 + cth * a1; }
      for (int k = 0; k < 3; ++k) { double v0 = V[k][p], v1 = V[k][q]; V[k][p] = cth * v0 - sth * v1; V[k][q] = sth * v0 + cth * v1; }
    }
  }
  double lam[3] = { A[0][0], A[1][1], A[2][2] };
  for (int i = 0; i < 2; ++i)
    for (int j = i + 1; j < 3; ++j)
      if (lam[j] > lam[i]) {
        double tl = lam[i]; lam[i] = lam[j]; lam[j] = tl;
        for (int k = 0; k < 3; ++k) { double tv = V[k][i]; V[k][i] = V[k][j]; V[k][j] = tv; }
      }
  double U[3][3];
  for (int c = 0; c < 3; ++c) {
    double w0 = H[0][0] * V[0][c] + H[0][1] * V[1][c] + H[0][2] * V[2][c];
    double w1 = H[1][0] * V[0][c] + H[1][1] * V[1][c] + H[1][2] * V[2][c];
    double w2 = H[2][0] * V[0][c] + H[2][1] * V[1][c] + H[2][2] * V[2][c];
    double n = sqrt(w0 * w0 + w1 * w1 + w2 * w2);
    if (n > 1e-150) { U[0][c] = w0 / n; U[1][c] = w1 / n; U[2][c] = w2 / n; }
    else if (c == 2) {
      U[0][2] = U[1][0] * U[2][1] - U[2][0] * U[1][1];
      U[1][2] = U[2][0] * U[0][1] - U[0][0] * U[2][1];
      U[2][2] = U[0][0] * U[1][1] - U[1][0] * U[0][1];
    } else { U[0][c] = (c == 0) ? 1.0 : 0.0; U[1][c] = (c == 1) ? 1.0 : 0.0; U[2][c] = 0.0; }
  }
  double R0[3][3];
  for (int i = 0; i < 3; ++i)
    for (int j = 0; j < 3; ++j)
      R0[i][j] = V[i][0] * U[j][0] + V[i][1] * U[j][1] + V[i][2] * U[j][2];
  double det = R0[0][0] * (R0[1][1] * R0[2][2] - R0[1][2] * R0[2][1])
             - R0[0][1] * (R0[1][0] * R0[2][2] - R0[1][2] * R0[2][0])
             + R0[0][2] * (R0[1][0] * R0[2][1] - R0[1][1] * R0[2][0]);
  double sgn = (det < 0.0) ? -1.0 : 1.0;
  for (int i = 0; i < 3; ++i)
    for (int j = 0; j < 3; ++j)
      R[i][j] = V[i][0] * U[j][0] + V[i][1] * U[j][1] + sgn * V[i][2] * U[j][2];
}

__global__ __launch_bounds__(TPB) void capr_final(const float* __restrict__ part16,
                                                  const float* __restrict__ partA,
                                                  float* __restrict__ out) {
  __shared__ float red[TPB];
  __shared__ float sums[16];
  __shared__ float losses[2];
  const int t = threadIdx.x;
  for (int q = 0; q < 16; ++q) {
    float s = 0.f;
    for (int i = t; i < NB; i += TPB) s += part16[i * 16 + q];
    red[t] = s; __syncthreads();
    for (int o = TPB / 2; o > 0; o >>= 1) { if (t < o) red[t] += red[t + o]; __syncthreads(); }
    if (t == 0) sums[q] = red[0];
    __syncthreads();
  }
  for (int q = 0; q < 2; ++q) {
    float s = 0.f;
    for (int i = t; i < NB; i += TPB) s += partA[i * 2 + q];
    red[t] = s; __syncthreads();
    for (int o = TPB / 2; o > 0; o >>= 1) { if (t < o) red[t] += red[t + o]; __syncthreads(); }
    if (t == 0) losses[q] = red[0];
    __syncthreads();
  }
  if (t == 0) {
    double Z = (double)sums[0];
    double c = 1.0 / (Z * (1.0 + 1e-6));      // w_i = e_i * c
    double s = 1.0 / (1.0 + 1e-6);            // sum w
    double cs[3], ct[3];
    for (int j = 0; j < 3; ++j) { cs[j] = c * (double)sums[1 + j]; ct[j] = c * (double)sums[4 + j]; }
    double H[3][3];
    for (int j = 0; j < 3; ++j)
      for (int k = 0; k < 3; ++k)
        H[j][k] = c * (double)sums[7 + j * 3 + k] + (s - 2.0) * cs[j] * ct[k] + ((j == k) ? 1e-6 : 0.0);
    double R[3][3];
    capr_kabsch3(H, R);
    for (int j = 0; j < 3; ++j)
      for (int k = 0; k < 3; ++k) out[j * 3 + k] = (float)R[j][k];
    for (int j = 0; j < 3; ++j)
      out[9 + j] = (float)(ct[j] - (R[j][0] * cs[0] + R[j][1] * cs[1] + R[j][2] * cs[2]));
    out[12] = (losses[0] + losses[1]) / (float)NPTS;   // rot.mean + feat.mean
  }
}

// =====================================================================
extern "C" void kernel_launch(void* const* d_in, const int* in_sizes, int n_in,
                              void* d_out, int out_size, void* d_ws, size_t ws_size,
                              hipStream_t stream) {
  (void)in_sizes; (void)n_in; (void)out_size; (void)ws_size;
  const float* hs = (const float*)d_in[0];
  const float* xs = (const float*)d_in[1];
  const float* ht = (const float*)d_in[4];
  const float* xt = (const float*)d_in[5];
  const float* labels = (const float*)d_in[9];
  const float* gt = (const float*)d_in[10];
  const float* W1 = (const float*)d_in[11];
  const float* b1 = (const float*)d_in[12];
  const float* W2 = (const float*)d_in[13];
  const float* b2 = (const float*)d_in[14];
  const float* W3 = (const float*)d_in[15];
  const float* b3 = (const float*)d_in[16];
  float* out = (float*)d_out;

  // workspace carve-up (~2.2 MB)
  float* sim = (float*)d_ws;                                        // N floats (8B aligned)
  unsigned long long* maxkey = (unsigned long long*)(sim + NPTS);   // 1
  unsigned* hist = (unsigned*)(maxkey + 1);                         // 256
  unsigned* sel = hist + 256;                                       // 8
  float* scal = (float*)(sel + 8);                                  // 8
  unsigned* topidx = (unsigned*)(scal + 8);                         // 128
  unsigned* eqbuf = topidx + 128;                                   // 4096
  float* partA = (float*)(eqbuf + 4096);                            // NB*2
  float* partS = partA + NB * 2;                                    // NB
  float* part16 = partS + NB;                                       // NB*16

  capr_init<<<1, 256, 0, stream>>>(hist, sel, maxkey);
  capr_copy<<<4096, TPB, 0, stream>>>(hs, xs, ht, xt, out);
  capr_point<<<NB, TPB, 0, stream>>>(hs, ht, xs, xt, labels, gt, sim, maxkey, partA);
  for (int p = 0; p < 4; ++p) {
    int shift = 24 - 8 * p;
    if (p > 0) capr_histzero<<<1, 256, 0, stream>>>(hist);
    capr_hist<<<NB, TPB, 0, stream>>>(sim, hist, sel, shift);
    capr_sel<<<1, 1, 0, stream>>>(hist, sel, topidx, maxkey, shift);
  }
  capr_gather<<<NB, TPB, 0, stream>>>(sim, sel, topidx, eqbuf);
  capr_fill<<<1, TPB, 0, stream>>>(sel, topidx, eqbuf);
  capr_mlp<<<1, TPB, 0, stream>>>(hs, ht, W1, b1, W2, b2, W3, b3, topidx, maxkey, scal);
  capr_scatter<<<1, KTOP, 0, stream>>>(sim, topidx, scal);
  capr_sum1<<<NB, TPB, 0, stream>>>(sim, partS, sel);
  capr_prep<<<1, TPB, 0, stream>>>(partS, sel, scal);
  capr_wsum<<<NB, TPB, 0, stream>>>(sim, xs, xt, scal, part16);
  capr_final<<<1, TPB, 0, stream>>>(part16, partA, out);
}